// BuildGCN_75909251989914
// MI455X (gfx1250) — compile-verified
//
#include <hip/hip_runtime.h>
#include <math.h>

typedef float v2f __attribute__((ext_vector_type(2)));
typedef float v8f __attribute__((ext_vector_type(8)));

#define NNODE 50000
#define NEDGE 800000
#define KP1   25000
#define KP2   12500

// ---------------- workspace arena (float offsets) ----------------
constexpr long long F_X1   = 0;
constexpr long long F_X2   = F_X1   + (long long)NNODE*128;
constexpr long long F_X3   = F_X2   + (long long)KP1*256;   // x3, later reused as xb
constexpr long long F_X2G  = F_X3   + (long long)KP2*512;
constexpr long long F_X3G  = F_X2G  + (long long)KP1*128;
constexpr long long F_AGG  = F_X3G  + (long long)KP2*256;
constexpr long long F_PRE  = F_AGG  + (long long)NNODE*128;
constexpr long long F_XU1  = F_PRE  + (long long)NNODE*128;
constexpr long long F_XU1U = F_XU1  + (long long)KP2*256;
constexpr long long F_XU2  = F_XU1U + (long long)KP1*256;
constexpr long long F_XU2U = F_XU2  + (long long)KP1*128;
constexpr long long F_XO   = F_XU2U + (long long)NNODE*128;
constexpr long long F_XO2  = F_XO   + (long long)NNODE*128;
constexpr long long F_WF   = F_XO2  + (long long)NNODE*128;
constexpr long long F_BF   = F_WF   + 256*128;
constexpr long long F_CSP  = F_BF   + 128;
constexpr long long F_CNT  = F_CSP  + 128;
constexpr long long F_SCORE= F_CNT  + (NNODE+1);
constexpr long long F_SUMS = F_SCORE+ NNODE;
constexpr long long F_MR   = F_SUMS + 1024;
constexpr long long F_INVN = F_MR   + 1024;
constexpr long long F_FEND = F_INVN + 8;
// int region (relative to int* base placed at F_FEND floats)
constexpr long long I_KEYS  = 0;
constexpr long long I_PERM1 = I_KEYS  + NNODE;
constexpr long long I_POS1  = I_PERM1 + KP1;
constexpr long long I_PERM2 = I_POS1  + (NNODE+1);
constexpr long long I_POS2  = I_PERM2 + KP2;
constexpr long long I_SRC2  = I_POS2  + (KP1+1);
constexpr long long I_DST2  = I_SRC2  + NEDGE;
constexpr long long I_SRC3  = I_DST2  + NEDGE;
constexpr long long I_DST3  = I_SRC3  + NEDGE;
constexpr long long I_SEL   = I_DST3  + NEDGE;

// ---------------- kernels ----------------

// c = irfft(wc, 128): spatial circular-conv kernel of the spectral branch
__global__ void k_cspat(const float* __restrict__ cw, float* __restrict__ csp) {
  int d = threadIdx.x;
  if (d >= 128) return;
  float acc = cw[0];                       // k=0: real part only (irfft drops imag)
  for (int k = 1; k < 64; ++k) {
    int t = (k * d) & 127;
    float ang = 6.28318530717958647692f * (float)t / 128.0f;
    acc += 2.0f * (cw[2*k] * cosf(ang) - cw[2*k+1] * sinf(ang));
  }
  acc += cw[128] * ((d & 1) ? -1.0f : 1.0f);  // k=64: cos(pi*d)
  csp[d] = acc * (1.0f / 128.0f);
}

// fold spectral branch into level-1 weight/bias:
//   Wf[r][m] = g1_W[r][m] + (r<128 ? alpha * sum_n g1f_W[r][n] * c[(m-n)%128] : 0)
//   bf[m]    = g1_b[m]    + alpha * sum_n g1f_b[n] * c[(m-n)%128]
__global__ void k_fold(const float* __restrict__ g1W, const float* __restrict__ g1b,
                       const float* __restrict__ gfW, const float* __restrict__ gfb,
                       const float* __restrict__ alpha, const float* __restrict__ csp,
                       float* __restrict__ Wf, float* __restrict__ bf) {
  int i = blockIdx.x * blockDim.x + threadIdx.x;
  float a = alpha[0];
  if (i < 256*128) {
    int r = i >> 7, m = i & 127;
    float v = g1W[i];
    if (r < 128) {
      float s = 0.0f;
      for (int n = 0; n < 128; ++n) s += gfW[r*128 + n] * csp[(m - n) & 127];
      v += a * s;
    }
    Wf[i] = v;
  } else if (i < 256*128 + 128) {
    int m = i - 256*128;
    float s = 0.0f;
    for (int n = 0; n < 128; ++n) s += gfb[n] * csp[(m - n) & 127];
    bf[m] = g1b[m] + a * s;
  }
}

__global__ void k_edge_count(const int* __restrict__ dst, int E, int num,
                             float* __restrict__ cnt) {
  int e = blockIdx.x * blockDim.x + threadIdx.x;
  if (e >= E) return;
  int d = dst[e];
  if (d < num) atomicAdd(&cnt[d], 1.0f);
}

__global__ void k_edge_agg(const float* __restrict__ x, const int* __restrict__ src,
                           const int* __restrict__ dst, long long total, int groups,
                           int num, int C, float* __restrict__ agg) {
  long long i = (long long)blockIdx.x * blockDim.x + threadIdx.x;
  if (i >= total) return;
  int e  = (int)(i / groups);
  int c4 = (int)(i % groups) << 2;
  int d = dst[e];
  if (d >= num) return;                 // dummy bucket dropped
  int s = src[e];
  const float4 v = *(const float4*)(x + (long long)s * C + c4);
  float* ap = agg + (long long)d * C + c4;
  atomicAdd(ap + 0, v.x); atomicAdd(ap + 1, v.y);
  atomicAdd(ap + 2, v.z); atomicAdd(ap + 3, v.w);
}

__global__ void k_agg_div(float* __restrict__ agg, const float* __restrict__ cnt,
                          long long total, int C) {
  long long i = (long long)blockIdx.x * blockDim.x + threadIdx.x;
  if (i >= total) return;
  int n = (int)(i / C);
  float cn = cnt[n];
  agg[i] *= (cn >= 1.0f) ? (1.0f / cn) : 1.0f;
}

// out[M,Co] = [A0 | A1] @ W + bias  ; split at K0. One wave per 16x16 tile,
// fp32 WMMA (V_WMMA_F32_16X16X4_F32), K step 4. Two pointer-increment phases:
// no per-iteration 64-bit address math, no split-K branch in the hot loop.
__global__ void __launch_bounds__(256)
k_gemm(const float* __restrict__ A0, const float* __restrict__ A1,
       int K0, int K, const float* __restrict__ W, const float* __restrict__ bias,
       float* __restrict__ out, int M, int Co, int tn, long long tiles) {
  int wave = (blockIdx.x << 3) + (threadIdx.x >> 5);
  if ((long long)wave >= tiles) return;
  int m0 = (wave / tn) << 4;
  int n0 = (wave % tn) << 4;
  int lane = threadIdx.x & 31;
  int mi = lane & 15;          // row within tile (A), col within tile (B/C)
  int kh = lane >> 4;          // K-half select
  int mrow = m0 + mi; if (mrow > M - 1) mrow = M - 1;   // clamp (store guarded)
  int col = n0 + mi;
  const long long wstep = (long long)Co * 4;
  v8f c = {};
  // phase 0: A0 (k = 0 .. K0)
  {
    const float* ap = A0 + (long long)mrow * K0 + (kh << 1);
    const float* wp = W + (long long)(kh << 1) * Co + col;
    for (int k = 0; k < K0; k += 4) {
      v2f a; a.x = ap[0]; a.y = ap[1];   // A 16x4: lanes0-15 K={0,1}, lanes16-31 K={2,3}
      v2f b; b.x = wp[0]; b.y = wp[Co];  // B 4x16: VGPR0 rows {0,2}, VGPR1 rows {1,3}
      c = __builtin_amdgcn_wmma_f32_16x16x4_f32(false, a, false, b, (short)0, c, false, false);
      ap += 4; wp += wstep;
    }
  }
  // phase 1: A1 (k = K0 .. K)
  int K1 = K - K0;
  if (K1 > 0) {
    const float* ap = A1 + (long long)mrow * K1 + (kh << 1);
    const float* wp = W + ((long long)K0 + (kh << 1)) * Co + col;
    for (int k = 0; k < K1; k += 4) {
      v2f a; a.x = ap[0]; a.y = ap[1];
      v2f b; b.x = wp[0]; b.y = wp[Co];
      c = __builtin_amdgcn_wmma_f32_16x16x4_f32(false, a, false, b, (short)0, c, false, false);
      ap += 4; wp += wstep;
    }
  }
  float bv = bias[col];
#pragma unroll
  for (int j = 0; j < 8; ++j) {          // C: VGPR j = rows j (lanes0-15), j+8 (lanes16-31)
    int row = m0 + j + (kh << 3);
    if (row < M) out[(long long)row * Co + col] = c[j] + bv;
  }
}

__global__ void k_col_stats(const float* __restrict__ x, int M, int C,
                            float* __restrict__ sums) {
  int r0 = blockIdx.x * 64;
  int rend = r0 + 64; if (rend > M) rend = M;
  for (int c = threadIdx.x; c < C; c += blockDim.x) {
    float s = 0.0f, s2 = 0.0f;
    for (int r = r0; r < rend; ++r) {
      float v = x[(long long)r * C + c];
      s += v; s2 += v * v;
    }
    atomicAdd(&sums[c], s);
    atomicAdd(&sums[C + c], s2);
  }
}

__global__ void k_stats_final(const float* __restrict__ sums, float invM, int C,
                              float* __restrict__ mr) {
  int c = blockIdx.x * blockDim.x + threadIdx.x;
  if (c >= C) return;
  float m = sums[c] * invM;
  float v = sums[C + c] * invM - m * m;
  mr[c] = m;
  mr[C + c] = rsqrtf(v + 1e-5f);
}

__global__ void k_bn_gelu(const float* __restrict__ pre, const float* __restrict__ mr,
                          const float* __restrict__ g, const float* __restrict__ bb,
                          float* __restrict__ out, long long total, int C) {
  long long i = (long long)blockIdx.x * blockDim.x + threadIdx.x;
  if (i >= total) return;
  int c = (int)(i % C);
  float y = (pre[i] - mr[c]) * mr[C + c] * g[c] + bb[c];
  out[i] = 0.5f * y * (1.0f + erff(y * 0.70710678118654752f));
}

__global__ void k_wnorm(const float* __restrict__ w, int C, float* __restrict__ invn) {
  __shared__ float sh[256];
  float s = 0.0f;
  for (int c = threadIdx.x; c < C; c += 256) { float v = w[c]; s += v * v; }
  sh[threadIdx.x] = s; __syncthreads();
  for (int o = 128; o > 0; o >>= 1) {
    if (threadIdx.x < o) sh[threadIdx.x] += sh[threadIdx.x + o];
    __syncthreads();
  }
  if (threadIdx.x == 0) invn[0] = rsqrtf(sh[0]);
}

__global__ void k_score(const float* __restrict__ x, const float* __restrict__ w,
                        const float* __restrict__ invn, int M, int C,
                        float* __restrict__ score, unsigned* __restrict__ keys) {
  int gid = blockIdx.x * blockDim.x + threadIdx.x;
  int node = gid >> 5, lane = gid & 31;
  if (node >= M) return;
  float s = 0.0f;
  for (int c = lane; c < C; c += 32) s += x[(long long)node * C + c] * w[c];
  for (int o = 16; o > 0; o >>= 1) s += __shfl_xor(s, o, 32);
  if (lane == 0) {
    float sc = tanhf(s * invn[0]);
    score[node] = sc;
    unsigned u = __float_as_uint(sc);
    keys[node] = (u & 0x80000000u) ? ~u : (u | 0x80000000u);
  }
}

// exact k-th-largest via 4-pass radix select; sel={thr, n_strict_gt, n_eq_take}
__global__ void k_radix_kth(const unsigned* __restrict__ keys, int n, int k,
                            int* __restrict__ sel) {
  __shared__ unsigned hist[256];
  __shared__ unsigned s_prefix, s_mask;
  __shared__ int s_k;
  int tid = threadIdx.x;
  if (tid == 0) { s_prefix = 0u; s_mask = 0u; s_k = k; }
  __syncthreads();
  for (int pass = 0; pass < 4; ++pass) {
    int shift = 24 - 8 * pass;
    hist[tid] = 0u;
    __syncthreads();
    unsigned pfx = s_prefix, msk = s_mask;
    for (int i = tid; i < n; i += 256) {
      unsigned kk = keys[i];
      if ((kk & msk) == pfx) atomicAdd(&hist[(kk >> shift) & 0xFFu], 1u);
    }
    __syncthreads();
    if (tid == 0) {
      int rem = s_k, b = 255;
      for (; b > 0; --b) {
        int h = (int)hist[b];
        if (rem <= h) break;
        rem -= h;
      }
      s_prefix |= ((unsigned)b) << shift;
      s_mask   |= 0xFFu << shift;
      s_k = rem;
    }
    __syncthreads();
  }
  if (tid == 0) { sel[0] = (int)s_prefix; sel[1] = k - s_k; sel[2] = s_k; }
}

__global__ void k_select(const unsigned* __restrict__ keys, const int* __restrict__ sel,
                         int* __restrict__ ctr, int M, int* __restrict__ pos,
                         int* __restrict__ perm) {
  int n = blockIdx.x * blockDim.x + threadIdx.x;
  if (n > M) return;
  if (n == M) { pos[M] = -1; return; }          // dummy bucket
  unsigned thr = (unsigned)sel[0];
  int n_gt = sel[1], need = sel[2];
  unsigned kk = keys[n];
  int idx = -1;
  if (kk > thr) idx = atomicAdd(&ctr[0], 1);
  else if (kk == thr) {
    int e = atomicAdd(&ctr[1], 1);
    if (e < need) idx = n_gt + e;
  }
  pos[n] = idx;
  if (idx >= 0) perm[idx] = n;
}

__global__ void k_gate(const float* __restrict__ x, const float* __restrict__ score,
                       const int* __restrict__ perm, long long total, int C,
                       float* __restrict__ out) {
  long long i = (long long)blockIdx.x * blockDim.x + threadIdx.x;
  if (i >= total) return;
  int r = (int)(i / C), c = (int)(i % C);
  int n = perm[r];
  out[i] = x[(long long)n * C + c] * score[n];
}

__global__ void k_relabel(const int* __restrict__ sin, const int* __restrict__ din,
                          const int* __restrict__ pos, int E, int knew,
                          int* __restrict__ sout, int* __restrict__ dout) {
  int e = blockIdx.x * blockDim.x + threadIdx.x;
  if (e >= E) return;
  int s2 = pos[sin[e]], d2 = pos[din[e]];
  bool valid = (s2 >= 0) && (d2 >= 0);
  sout[e] = valid ? s2 : 0;
  dout[e] = valid ? d2 : knew;
}

__global__ void k_unpool(const float* __restrict__ xp, const float* __restrict__ xs,
                         const int* __restrict__ pos, long long total, int C,
                         float* __restrict__ out) {
  long long i = (long long)blockIdx.x * blockDim.x + threadIdx.x;
  if (i >= total) return;
  int n = (int)(i / C), c = (int)(i % C);
  float v = xs[i];
  int p = pos[n];
  if (p >= 0) v += xp[(long long)p * C + c];
  out[i] = v;
}

__global__ void k_fc(const float* __restrict__ x, const float* __restrict__ W,
                     const float* __restrict__ b, int M, float* __restrict__ out) {
  int i = blockIdx.x * blockDim.x + threadIdx.x;
  if (i >= M * 3) return;
  int n = i / 3, o = i % 3;
  float s = b[o];
  for (int c = 0; c < 128; ++c) s += x[(long long)n * 128 + c] * W[c * 3 + o];
  out[i] = s;
}

// ---------------- host orchestration ----------------
static inline unsigned nblk(long long t) { return (unsigned)((t + 255) / 256); }

extern "C" void kernel_launch(void* const* d_in, const int* in_sizes, int n_in,
                              void* d_out, int out_size, void* d_ws, size_t ws_size,
                              hipStream_t stream) {
  (void)in_sizes; (void)n_in; (void)out_size; (void)ws_size;

  const float* ndata = (const float*)d_in[0];
  const int*   edata = (const int*)d_in[1];
  const int* src0 = edata;
  const int* dst0 = edata + NEDGE;
  const float* g1_W  = (const float*)d_in[2];
  const float* g1_b  = (const float*)d_in[3];
  const float* g1f_W = (const float*)d_in[4];
  const float* g1f_b = (const float*)d_in[5];
  const float* g1_cw = (const float*)d_in[6];
  const float* alpha = (const float*)d_in[7];
  const float* bn1_g = (const float*)d_in[8];
  const float* bn1_b = (const float*)d_in[9];
  const float* p1_w  = (const float*)d_in[10];
  const float* g2_W  = (const float*)d_in[11];
  const float* g2_b  = (const float*)d_in[12];
  const float* bn2_g = (const float*)d_in[13];
  const float* bn2_b = (const float*)d_in[14];
  const float* p2_w  = (const float*)d_in[15];
  const float* g3_W  = (const float*)d_in[16];
  const float* g3_b  = (const float*)d_in[17];
  const float* bn3_g = (const float*)d_in[18];
  const float* bn3_b = (const float*)d_in[19];
  const float* gb_W  = (const float*)d_in[20];
  const float* gb_b  = (const float*)d_in[21];
  const float* bnb_g = (const float*)d_in[22];
  const float* bnb_b = (const float*)d_in[23];
  const float* d1_W  = (const float*)d_in[24];
  const float* d1_b  = (const float*)d_in[25];
  const float* bnd1_g= (const float*)d_in[26];
  const float* bnd1_b= (const float*)d_in[27];
  const float* d2_W  = (const float*)d_in[28];
  const float* d2_b  = (const float*)d_in[29];
  const float* bnd2_g= (const float*)d_in[30];
  const float* bnd2_b= (const float*)d_in[31];
  const float* d3_W  = (const float*)d_in[32];
  const float* d3_b  = (const float*)d_in[33];
  const float* bnd3_g= (const float*)d_in[34];
  const float* bnd3_b= (const float*)d_in[35];
  const float* ls_W  = (const float*)d_in[36];
  const float* ls_b  = (const float*)d_in[37];
  const float* bnl_g = (const float*)d_in[38];
  const float* bnl_b = (const float*)d_in[39];
  const float* fc_W  = (const float*)d_in[40];
  const float* fc_b  = (const float*)d_in[41];

  float* WSF = (float*)d_ws;
  float* X1   = WSF + F_X1;
  float* X2   = WSF + F_X2;
  float* X3   = WSF + F_X3;
  float* X2G  = WSF + F_X2G;
  float* X3G  = WSF + F_X3G;
  float* AGG  = WSF + F_AGG;
  float* PRE  = WSF + F_PRE;
  float* XU1  = WSF + F_XU1;
  float* XU1U = WSF + F_XU1U;
  float* XU2  = WSF + F_XU2;
  float* XU2U = WSF + F_XU2U;
  float* XO   = WSF + F_XO;
  float* XO2  = WSF + F_XO2;
  float* WF   = WSF + F_WF;
  float* BF   = WSF + F_BF;
  float* CSP  = WSF + F_CSP;
  float* CNT  = WSF + F_CNT;
  float* SCORE= WSF + F_SCORE;
  float* SUMS = WSF + F_SUMS;
  float* MR   = WSF + F_MR;
  float* INVN = WSF + F_INVN;
  int* IB     = (int*)(WSF + F_FEND);
  unsigned* KEYS = (unsigned*)(IB + I_KEYS);
  int* PERM1 = IB + I_PERM1;
  int* POS1  = IB + I_POS1;
  int* PERM2 = IB + I_PERM2;
  int* POS2  = IB + I_POS2;
  int* SRC2  = IB + I_SRC2;
  int* DST2  = IB + I_DST2;
  int* SRC3  = IB + I_SRC3;
  int* DST3  = IB + I_DST3;
  int* SEL   = IB + I_SEL;

  auto mp = [&](const float* x, const int* s, const int* d, int num, int C) {
    hipMemsetAsync(AGG, 0, (size_t)num * C * sizeof(float), stream);
    hipMemsetAsync(CNT, 0, (size_t)num * sizeof(float), stream);
    k_edge_count<<<nblk(NEDGE), 256, 0, stream>>>(d, NEDGE, num, CNT);
    int groups = C >> 2;
    long long tot = (long long)NEDGE * groups;
    k_edge_agg<<<nblk(tot), 256, 0, stream>>>(x, s, d, tot, groups, num, C, AGG);
    long long tot2 = (long long)num * C;
    k_agg_div<<<nblk(tot2), 256, 0, stream>>>(AGG, CNT, tot2, C);
  };
  auto gemm = [&](const float* A0, const float* A1, int K0, int K,
                  const float* Wt, const float* bias, float* out, int M, int Co) {
    int tm = (M + 15) >> 4, tn = Co >> 4;
    long long tiles = (long long)tm * tn;
    k_gemm<<<(unsigned)((tiles + 7) / 8), 256, 0, stream>>>(A0, A1, K0, K, Wt, bias,
                                                            out, M, Co, tn, tiles);
  };
  auto bn = [&](const float* pre, const float* g, const float* bb, float* out,
                int M, int C) {
    hipMemsetAsync(SUMS, 0, 2 * C * sizeof(float), stream);
    k_col_stats<<<(unsigned)(((long long)M + 63) / 64), 256, 0, stream>>>(pre, M, C, SUMS);
    k_stats_final<<<nblk(C), 256, 0, stream>>>(SUMS, 1.0f / (float)M, C, MR);
    long long tot = (long long)M * C;
    k_bn_gelu<<<nblk(tot), 256, 0, stream>>>(pre, MR, g, bb, out, tot, C);
  };
  auto pool = [&](const float* x, int M, int C, int kkeep, const float* pw,
                  int* pos, int* perm, float* xg,
                  const int* sin, const int* din, int* sout, int* dout) {
    k_wnorm<<<1, 256, 0, stream>>>(pw, C, INVN);
    k_score<<<nblk((long long)M * 32), 256, 0, stream>>>(x, pw, INVN, M, C, SCORE, KEYS);
    k_radix_kth<<<1, 256, 0, stream>>>(KEYS, M, kkeep, SEL);
    hipMemsetAsync(SEL + 3, 0, 2 * sizeof(int), stream);
    k_select<<<nblk(M + 1), 256, 0, stream>>>(KEYS, SEL, SEL + 3, M, pos, perm);
    long long tot = (long long)kkeep * C;
    k_gate<<<nblk(tot), 256, 0, stream>>>(x, SCORE, perm, tot, C, xg);
    k_relabel<<<nblk(NEDGE), 256, 0, stream>>>(sin, din, pos, NEDGE, kkeep, sout, dout);
  };

  // level-1 weight folding (spectral branch -> circulant -> merged into g1_W)
  k_cspat<<<1, 128, 0, stream>>>(g1_cw, CSP);
  k_fold<<<nblk(256 * 128 + 128), 256, 0, stream>>>(g1_W, g1_b, g1f_W, g1f_b, alpha, CSP, WF, BF);

  // encoder level 1
  mp(ndata, src0, dst0, NNODE, 128);
  gemm(ndata, AGG, 128, 256, WF, BF, PRE, NNODE, 128);
  bn(PRE, bn1_g, bn1_b, X1, NNODE, 128);
  // pool 1
  pool(X1, NNODE, 128, KP1, p1_w, POS1, PERM1, X2G, src0, dst0, SRC2, DST2);
  // gcn 2
  mp(X2G, SRC2, DST2, KP1, 128);
  gemm(X2G, AGG, 128, 256, g2_W, g2_b, PRE, KP1, 256);
  bn(PRE, bn2_g, bn2_b, X2, KP1, 256);
  // pool 2
  pool(X2, KP1, 256, KP2, p2_w, POS2, PERM2, X3G, SRC2, DST2, SRC3, DST3);
  // gcn 3
  mp(X3G, SRC3, DST3, KP2, 256);
  gemm(X3G, AGG, 256, 512, g3_W, g3_b, PRE, KP2, 512);
  bn(PRE, bn3_g, bn3_b, X3, KP2, 512);
  // bottleneck (xb overwrites X3 after its GEMM has consumed it)
  mp(X3, SRC3, DST3, KP2, 512);
  gemm(X3, AGG, 512, 1024, gb_W, gb_b, PRE, KP2, 512);
  bn(PRE, bnb_g, bnb_b, X3, KP2, 512);
  // decoder 1
  mp(X3, SRC3, DST3, KP2, 512);
  gemm(X3, AGG, 512, 1024, d1_W, d1_b, PRE, KP2, 256);
  bn(PRE, bnd1_g, bnd1_b, XU1, KP2, 256);
  k_unpool<<<nblk((long long)KP1 * 256), 256, 0, stream>>>(XU1, X2, POS2,
                                                           (long long)KP1 * 256, 256, XU1U);
  // decoder 2
  mp(XU1U, SRC2, DST2, KP1, 256);
  gemm(XU1U, AGG, 256, 512, d2_W, d2_b, PRE, KP1, 128);
  bn(PRE, bnd2_g, bnd2_b, XU2, KP1, 128);
  k_unpool<<<nblk((long long)NNODE * 128), 256, 0, stream>>>(XU2, X1, POS1,
                                                             (long long)NNODE * 128, 128, XU2U);
  // decoder 3
  mp(XU2U, src0, dst0, NNODE, 128);
  gemm(XU2U, AGG, 128, 256, d3_W, d3_b, PRE, NNODE, 128);
  bn(PRE, bnd3_g, bnd3_b, XO, NNODE, 128);
  // stacked linear + final
  gemm(XO, XO, 128, 128, ls_W, ls_b, PRE, NNODE, 128);
  bn(PRE, bnl_g, bnl_b, XO2, NNODE, 128);
  k_fc<<<nblk((long long)NNODE * 3), 256, 0, stream>>>(XO2, fc_W, fc_b, NNODE, (float*)d_out);
}